// LUTBlock_21878563405942
// MI455X (gfx1250) — compile-verified
//
#include <hip/hip_runtime.h>

// Problem constants (match reference)
#define BN   16384   // batch
#define IND  1024    // in features
#define OUTD 512     // out features
#define TT   64      // tables
#define CC   8       // comparisons per table
#define RR   256     // rows per table (2^CC)

// Tiling for the gather kernel
#define OC   64      // out-columns per workgroup
#define BT   512     // batch rows per workgroup
#define QX   16      // threadIdx.x: float4 quads over OC (16*4 = 64 floats)
#define SY   64      // threadIdx.y: row slots
#define RPT  8       // rows per thread (SY * RPT = BT)

// ---------------------------------------------------------------------------
// CDNA5 async global->LDS copy (ASYNCcnt-tracked). Inline asm is portable
// across ROCm7.2 / amdgpu-toolchain (builtin arity differs between them).
// VDST = 32-bit LDS byte offset VGPR, VADDR = 64-bit global address pair.
// ---------------------------------------------------------------------------
__device__ __forceinline__ void async_ld_b128(unsigned lds_off, const float* gaddr) {
  asm volatile("global_load_async_to_lds_b128 %0, %1, off"
               :: "v"(lds_off), "v"(gaddr)
               : "memory");
}

__device__ __forceinline__ void wait_async0() {
#if defined(__has_builtin) && __has_builtin(__builtin_amdgcn_s_wait_asynccnt)
  __builtin_amdgcn_s_wait_asynccnt(0);
#else
  asm volatile("s_wait_asynccnt 0" ::: "memory");
#endif
}

// ---------------------------------------------------------------------------
// Kernel 1: per (batch row, table) 8-bit sign-pattern index.
// idx8 layout: [T][B] (transposed) so the gather kernel reads it coalesced.
// ---------------------------------------------------------------------------
__global__ __launch_bounds__(256)
void lut_idx_kernel(const float* __restrict__ x,
                    const int*   __restrict__ anchors_a,
                    const int*   __restrict__ anchors_b,
                    unsigned char* __restrict__ idx8) {
  __shared__ int sa[TT * CC];
  __shared__ int sb[TT * CC];
  for (int i = threadIdx.x; i < TT * CC; i += 256) {
    sa[i] = anchors_a[i];
    sb[i] = anchors_b[i];
  }
  __syncthreads();

  const int b = blockIdx.x * 256 + threadIdx.x;
  const float* xr = x + (size_t)b * IND;
  for (int t = 0; t < TT; ++t) {
    unsigned u = 0;
#pragma unroll
    for (int c = 0; c < CC; ++c) {
      const float d = xr[sa[t * CC + c]] - xr[sb[t * CC + c]];
      u |= (d > 0.0f) ? (1u << c) : 0u;
    }
    idx8[(size_t)t * BN + b] = (unsigned char)u;
  }
}

// ---------------------------------------------------------------------------
// Kernel 2: LDS-staged table gather + accumulate.
// Per workgroup: BT=512 batch rows x OC=64 out cols. Double-buffered 64 KB
// table slices filled with async-to-LDS copies; gathers served by ds_load_b128.
// Indices staged permuted so each thread fetches its 8 per-table indices with
// one ds_load_b64. Total LDS = 160 KB -> two workgroups can share a WGP.
// ---------------------------------------------------------------------------
__global__ __launch_bounds__(1024, 1)
void lut_gather_kernel(const float* __restrict__ table,
                       const unsigned char* __restrict__ idx8,
                       float* __restrict__ y) {
  __shared__ __align__(16) float buf[2][RR * OC];          // 2 x 64 KB slices
  // permuted: s_idx[t][y][k] = idx of global row (bBase + y + k*SY)
  __shared__ __align__(16) unsigned char s_idx[TT * BT];   // 32 KB indices

  const int ftid   = threadIdx.y * QX + threadIdx.x;       // 0..1023
  const int ocBase = blockIdx.y * OC;
  const int bBase  = blockIdx.x * BT;

  // ---- prologue: async-fill slice 0 into buf[0] (overlaps idx staging) ----
  {
    const float* gbase = table + ocBase;
    const unsigned ldsBase = (unsigned)(unsigned long long)(&buf[0][0]);
#pragma unroll
    for (int i = 0; i < 4; ++i) {
      const int p  = ftid + i * 1024;      // 16B chunk id, 0..4095 (64 KB)
      const int r  = p >> 4;               // table row (256 B per row)
      const int cq = p & 15;               // chunk within row
      async_ld_b128(ldsBase + (unsigned)p * 16u,
                    gbase + (size_t)r * OUTD + cq * 4);
    }
  }

  // ---- stage this tile's indices, permuted for per-thread b64 reads ----
  {
#pragma unroll
    for (int i = 0; i < 2; ++i) {
      const int vi = ftid + i * 1024;      // 0..2047 (16B vectors)
      const int t  = vi >> 5;              // 32 vectors (512 B) per table
      const int j  = vi & 31;
      const uint4 v =
          *(const uint4*)(idx8 + (size_t)t * BN + bBase + (size_t)j * 16);
      const unsigned char* pb = (const unsigned char*)&v;
#pragma unroll
      for (int e = 0; e < 16; ++e) {
        const int local = j * 16 + e;                  // row within tile 0..511
        const int yy = local & (SY - 1);
        const int kk = local >> 6;                      // /SY
        s_idx[t * BT + yy * RPT + kk] = pb[e];
      }
    }
  }
  wait_async0();
  __syncthreads();

  float4 acc[RPT];
#pragma unroll
  for (int k = 0; k < RPT; ++k) acc[k] = make_float4(0.f, 0.f, 0.f, 0.f);

  for (int t = 0; t < TT; ++t) {
    const int cur = t & 1;

    // issue async fill of next slice (overlaps with gather below)
    if (t + 1 < TT) {
      const float* gbase = table + (size_t)(t + 1) * RR * OUTD + ocBase;
      const unsigned ldsBase = (unsigned)(unsigned long long)(&buf[cur ^ 1][0]);
#pragma unroll
      for (int i = 0; i < 4; ++i) {
        const int p  = ftid + i * 1024;
        const int r  = p >> 4;
        const int cq = p & 15;
        async_ld_b128(ldsBase + (unsigned)p * 16u,
                      gbase + (size_t)r * OUTD + cq * 4);
      }
    }

    // one b64 LDS read fetches this thread's 8 row indices for table t
    const unsigned long long iw =
        *(const unsigned long long*)(&s_idx[t * BT + threadIdx.y * RPT]);

    // gather + accumulate from current slice (LDS)
    const float* sbuf = &buf[cur][0];
#pragma unroll
    for (int k = 0; k < RPT; ++k) {
      const int r = (int)((iw >> (8 * k)) & 0xFFull);    // table row 0..255
      const float4 v =
          *(const float4*)(sbuf + r * OC + threadIdx.x * 4);
      acc[k].x += v.x; acc[k].y += v.y; acc[k].z += v.z; acc[k].w += v.w;
    }

    if (t + 1 < TT) wait_async0();   // next slice resident before next iter
    __syncthreads();                  // all waves done reading buf[cur]
  }

  // ---- store: fully coalesced float4 ----
#pragma unroll
  for (int k = 0; k < RPT; ++k) {
    const int g = bBase + threadIdx.y + k * SY;
    *(float4*)(y + (size_t)g * OUTD + ocBase + threadIdx.x * 4) = acc[k];
  }
}

// ---------------------------------------------------------------------------
extern "C" void kernel_launch(void* const* d_in, const int* in_sizes, int n_in,
                              void* d_out, int out_size, void* d_ws, size_t ws_size,
                              hipStream_t stream) {
  const float* x       = (const float*)d_in[0];
  const float* table   = (const float*)d_in[1];
  const int*   aa      = (const int*)d_in[2];
  const int*   bb      = (const int*)d_in[3];
  float*       y       = (float*)d_out;
  unsigned char* idx8  = (unsigned char*)d_ws;   // T*B = 1 MB scratch

  lut_idx_kernel<<<dim3(BN / 256), dim3(256), 0, stream>>>(x, aa, bb, idx8);

  lut_gather_kernel<<<dim3(BN / BT, OUTD / OC), dim3(QX, SY), 0, stream>>>(
      table, idx8, y);
}